// SingleExpertModel_30288109372027
// MI455X (gfx1250) — compile-verified
//
#include <hip/hip_runtime.h>

// Fused segment-mean + projection for:
//   h_words = segment_mean(h_sub, word_ids)   (B, MW, D)
//   logits  = h_words @ W + b                 (B, MW, L)
// Using linearity: mean(h) @ W == mean(h @ W), so project each subtoken to
// L=8 with V_WMMA_F32_16X16X4_F32, then segment-reduce only 8 floats/token.
// Single streaming pass over h_sub (100.7 MB) -> bandwidth bound (~4.3us at
// 23.3 TB/s on MI455X).

typedef __attribute__((ext_vector_type(2))) float v2f;
typedef __attribute__((ext_vector_type(8))) float v8f;

#define B_      64
#define S_      512
#define D_      768
#define MW_     256
#define L_      8
#define CHUNKS  4
#define CHUNK_S (S_ / CHUNKS)          // 128 tokens per block
#define THREADS 256                    // 8 waves (wave32)
#define PART_STRIDE (MW_ * (L_ + 1))   // 2304 floats per (batch, chunk) partial

// ---------------------------------------------------------------------------
// Kernel 1: per (batch, chunk-of-128-tokens) block.
//   - each wave owns a 16-token tile, K-loop of 192 x v_wmma_f32_16x16x4_f32
//   - accumulate projected tokens into LDS segment sums/counts (ds_add_f32)
//   - flush partials to workspace
// ---------------------------------------------------------------------------
__global__ __launch_bounds__(THREADS)
void seg_proj_wmma_kernel(const float* __restrict__ h,    // (B, S, D)
                          const int*   __restrict__ wid,  // (B, S)
                          const float* __restrict__ W,    // (D, L)
                          float* __restrict__ part)       // (B*CHUNKS, PART_STRIDE)
{
    __shared__ float Wt[L_ * D_];       // Wt[n*D_ + k] = W[k*L_ + n]
    __shared__ float seg[PART_STRIDE];  // [0 .. MW_*L_) sums, then MW_ counts

    const int tid  = threadIdx.x;
    const int b    = blockIdx.x / CHUNKS;
    const int c    = blockIdx.x % CHUNKS;
    const int wave = tid >> 5;
    const int lane = tid & 31;

    // Zero segment accumulators.
    for (int i = tid; i < PART_STRIDE; i += THREADS) seg[i] = 0.0f;
    // Stage W transposed into LDS (24 KB), so B-fragment loads are ds_load_b64.
    for (int i = tid; i < L_ * D_; i += THREADS) {
        const int n = i / D_;
        const int k = i - n * D_;
        Wt[i] = W[k * L_ + n];
    }
    __syncthreads();

    // WMMA f32 16x16x4 operand layout (wave32):
    //   A (16x4, MxK):  lanes 0-15 -> M=lane,    K = {0,1}; lanes 16-31 -> K = {2,3}
    //   B (4x16, KxN):  lanes 0-15 -> N=lane,    K = {0,1}; lanes 16-31 -> K = {2,3}
    //   D (16x16):      VGPR v, lane -> M = v + 8*(lane/16), N = lane%16
    const int tok0 = c * CHUNK_S + wave * 16;  // this wave's 16-token tile
    const int m    = lane & 15;
    const int khi  = (lane >> 4) * 2;

    const float* __restrict__ arow = h + (size_t)(b * S_ + tok0 + m) * D_ + khi;
    // Lanes with N>=8 replicate N&7; D columns 8..15 are never read.
    const float* __restrict__ brow = Wt + (lane & 7) * D_ + khi;

    v8f acc = {};
#pragma unroll 4
    for (int k0 = 0; k0 < D_; k0 += 4) {
        v2f a  = { arow[k0], arow[k0 + 1] };
        v2f w2 = { brow[k0], brow[k0 + 1] };
        acc = __builtin_amdgcn_wmma_f32_16x16x4_f32(
            /*neg_a=*/false, a, /*neg_b=*/false, w2,
            /*c_mod=*/(short)0, acc, /*reuse_a=*/false, /*reuse_b=*/false);
    }

    // Scatter the 16x8 projected tile into LDS segment accumulators.
    const int n = lane & 15;
    if (n < L_) {
        const int mbase = (lane >> 4) * 8;
#pragma unroll
        for (int v = 0; v < 8; ++v) {
            const int tok = tok0 + mbase + v;
            const int w   = wid[b * S_ + tok];
            if (w >= 0 && w < MW_) {                 // invalid ids: overflow bucket -> skip
                atomicAdd(&seg[w * L_ + n], acc[v]); // ds_add_f32
                if (n == 0) atomicAdd(&seg[MW_ * L_ + w], 1.0f);
            }
        }
    }
    __syncthreads();

    float* __restrict__ out = part + (size_t)blockIdx.x * PART_STRIDE;
    for (int i = tid; i < PART_STRIDE; i += THREADS) out[i] = seg[i];
}

// ---------------------------------------------------------------------------
// Kernel 2: reduce the CHUNKS partials per (batch, word), divide by count,
// add bias. Zero-count words output exactly bias (matches reference).
// ---------------------------------------------------------------------------
__global__ __launch_bounds__(THREADS)
void seg_finalize_kernel(const float* __restrict__ part,
                         const float* __restrict__ bias,
                         float* __restrict__ out)
{
    const int t = blockIdx.x * blockDim.x + threadIdx.x;  // one per (b, word)
    if (t >= B_ * MW_) return;
    const int b = t / MW_;
    const int w = t - b * MW_;

    float s[L_] = {0.f, 0.f, 0.f, 0.f, 0.f, 0.f, 0.f, 0.f};
    float cnt = 0.f;
    for (int c = 0; c < CHUNKS; ++c) {
        const float* __restrict__ p = part + (size_t)(b * CHUNKS + c) * PART_STRIDE;
#pragma unroll
        for (int l = 0; l < L_; ++l) s[l] += p[w * L_ + l];
        cnt += p[MW_ * L_ + w];
    }
    const float inv = (cnt > 0.f) ? (1.0f / cnt) : 0.0f;
    float* __restrict__ o = out + (size_t)t * L_;
#pragma unroll
    for (int l = 0; l < L_; ++l) o[l] = bias[l] + s[l] * inv;
}

// ---------------------------------------------------------------------------
extern "C" void kernel_launch(void* const* d_in, const int* in_sizes, int n_in,
                              void* d_out, int out_size, void* d_ws, size_t ws_size,
                              hipStream_t stream)
{
    const float* h    = (const float*)d_in[0];  // (B,S,D) f32
    const int*   wid  = (const int*)  d_in[1];  // (B,S)   i32
    const float* W    = (const float*)d_in[2];  // (D,L)   f32
    const float* bias = (const float*)d_in[3];  // (L,)    f32
    // d_in[4] = max_words scalar (256), baked in as MW_.

    float* part = (float*)d_ws;  // (B*CHUNKS) * PART_STRIDE floats = 2.25 MB

    seg_proj_wmma_kernel<<<B_ * CHUNKS, THREADS, 0, stream>>>(h, wid, W, part);
    seg_finalize_kernel<<<(B_ * MW_ + THREADS - 1) / THREADS, THREADS, 0, stream>>>(
        part, bias, (float*)d_out);
}